// AlphaCompositor_53592601919501
// MI455X (gfx1250) — compile-verified
//
#include <hip/hip_runtime.h>
#include <stdint.h>

// AlphaCompositor for MI455X (gfx1250).
// fragments (4,8,256,256) i32, alphas (4,8,256,256) f32, ptclds (32,200000) f32
//   -> out (4,32,256,256) f32
// Memory-bound (~75 MB HBM, ptclds L2-resident). Uses CDNA5 async global->LDS
// staging + s_wait_asynccnt, transposed feature layout for 128B gathers, and
// non-temporal output stores to protect L2 residency of the point cloud.

constexpr int N_ = 4, K_ = 8, H_ = 256, W_ = 256, C_ = 32, P_ = 200000;
constexpr int PIX = 128;       // pixels per block tile (W_ % PIX == 0)
constexpr int THREADS = 256;   // 8 waves (wave32)

// ---- CDNA5 async global->LDS copy (tracked by ASYNCcnt) -------------------
__device__ __forceinline__ void async_copy_b128(uint32_t lds_off, uint64_t gaddr) {
#if __HIP_DEVICE_COMPILE__
  asm volatile("global_load_async_to_lds_b128 %0, %1, off"
               :: "v"(lds_off), "v"(gaddr) : "memory");
#endif
}
__device__ __forceinline__ void wait_async_zero() {
#if __HIP_DEVICE_COMPILE__
  asm volatile("s_wait_asynccnt 0" ::: "memory");
#endif
}

// ---- ptclds (C,P) -> (P,C): each point's 32 feats become one 128B row -----
__global__ __launch_bounds__(256)
void transpose_feats(const float* __restrict__ src, float* __restrict__ dst) {
  __shared__ float tile[C_][64 + 1];           // +1 pad: conflict-free
  const int p0 = blockIdx.x * 64;
  const int t  = threadIdx.x;
  const int lp = t & 63, lc0 = t >> 6;         // coalesced along P
#pragma unroll
  for (int i = 0; i < 8; ++i) {
    const int c = lc0 + i * 4;
    tile[c][lp] = __builtin_nontemporal_load(src + (size_t)c * P_ + p0 + lp);
  }
  __syncthreads();
  const int c = t & 31, pl0 = t >> 5;          // 128B store per wave
#pragma unroll
  for (int i = 0; i < 8; ++i) {
    const int pl = pl0 + i * 8;
    dst[(size_t)(p0 + pl) * C_ + c] = tile[c][pl];
  }
}

// ---- main compositing kernel ----------------------------------------------
template <bool TR>
__global__ __launch_bounds__(THREADS)
void composite(const int* __restrict__ frag, const float* __restrict__ alpha,
               const float* __restrict__ feats,  // TR ? (P,C) : (C,P)
               float* __restrict__ out) {
  __shared__ __align__(16) int   s_idx[K_][PIX];
  __shared__ __align__(16) float s_alp[K_][PIX];

  constexpr int TPR = W_ / PIX;                // tiles per image row
  const int tile = blockIdx.x;
  const int n    = tile / (H_ * TPR);
  const int rem  = tile % (H_ * TPR);
  const int h    = rem / TPR;
  const int w0   = (rem % TPR) * PIX;
  const int t    = threadIdx.x;

  { // Phase A: async-stage the (K x PIX) frag/alpha tiles into LDS.
    // 256 threads x b128 covers the 1024 elements of each tensor exactly.
    const int k = t >> 5;                      // 0..7
    const int w = (t & 31) * 4;                // 0,4,...,124  (16B aligned)
    const size_t g = ((size_t)(n * K_ + k) * H_ + h) * W_ + w0 + w;
    async_copy_b128((uint32_t)(uintptr_t)&s_idx[k][w],
                    (uint64_t)(uintptr_t)(frag + g));
    async_copy_b128((uint32_t)(uintptr_t)&s_alp[k][w],
                    (uint64_t)(uintptr_t)(alpha + g));
    wait_async_zero();
  }
  __syncthreads();

  // Phase B: each thread owns (pixel w, 16-channel half c0). Both threads of
  // a pixel redundantly compute the 8-step exclusive-transmittance scan in
  // registers (cheaper than a barrier + LDS write-back round trip).
  const int w  = t & (PIX - 1);
  const int c0 = (t >> 7) * 16;

  float wt[K_]; int idx[K_];
  {
    float T = 1.0f;
#pragma unroll
    for (int k = 0; k < K_; ++k) {
      const int f  = s_idx[k][w];
      float a      = s_alp[k][w];
      const bool v = (f >= 0);
      a = v ? a : 0.0f;
      wt[k]  = a * T;                          // w_k = a_k * prod_{j<k}(1-a_j)
      T     *= 1.0f - a;
      idx[k] = v ? f : 0;                      // safe gather index
    }
  }

  float acc[16];
#pragma unroll
  for (int j = 0; j < 16; ++j) acc[j] = 0.0f;

  if constexpr (TR) {
#pragma unroll
    for (int k = 0; k < K_; ++k) {
      const float4* fp = (const float4*)(feats + (size_t)idx[k] * C_ + c0);
      const float4 f0 = fp[0], f1 = fp[1], f2 = fp[2], f3 = fp[3];
      const float wk = wt[k];
      acc[0]  += wk * f0.x; acc[1]  += wk * f0.y; acc[2]  += wk * f0.z; acc[3]  += wk * f0.w;
      acc[4]  += wk * f1.x; acc[5]  += wk * f1.y; acc[6]  += wk * f1.z; acc[7]  += wk * f1.w;
      acc[8]  += wk * f2.x; acc[9]  += wk * f2.y; acc[10] += wk * f2.z; acc[11] += wk * f2.w;
      acc[12] += wk * f3.x; acc[13] += wk * f3.y; acc[14] += wk * f3.z; acc[15] += wk * f3.w;
    }
  } else { // fallback: gather from original (C,P) layout
#pragma unroll
    for (int k = 0; k < K_; ++k) {
      const float wk = wt[k];
      const float* base = feats + idx[k];
#pragma unroll
      for (int j = 0; j < 16; ++j) acc[j] += wk * base[(size_t)(c0 + j) * P_];
    }
  }

  // NCHW stores: contiguous in w across lanes, NT to keep L2 for ptclds.
  float* op = out + ((size_t)(n * C_ + c0) * (H_ * W_)) + (size_t)h * W_ + w0 + w;
#pragma unroll
  for (int j = 0; j < 16; ++j)
    __builtin_nontemporal_store(acc[j], op + (size_t)j * (H_ * W_));
}

extern "C" void kernel_launch(void* const* d_in, const int* in_sizes, int n_in,
                              void* d_out, int out_size, void* d_ws, size_t ws_size,
                              hipStream_t stream) {
  (void)in_sizes; (void)n_in; (void)out_size;
  const int*   frag  = (const int*)d_in[0];
  const float* alpha = (const float*)d_in[1];
  const float* ptc   = (const float*)d_in[2];
  float* out = (float*)d_out;

  const size_t needT  = (size_t)P_ * C_ * sizeof(float);
  const int    blocks = N_ * H_ * (W_ / PIX);   // 2048

  if (d_ws && ws_size >= needT) {
    transpose_feats<<<P_ / 64, 256, 0, stream>>>(ptc, (float*)d_ws);
    composite<true><<<blocks, THREADS, 0, stream>>>(frag, alpha,
                                                    (const float*)d_ws, out);
  } else {
    composite<false><<<blocks, THREADS, 0, stream>>>(frag, alpha, ptc, out);
  }
}